// ManualLoRALayer_30803505447138
// MI455X (gfx1250) — compile-verified
//
#include <hip/hip_runtime.h>
#include <hip/hip_bf16.h>

typedef __attribute__((ext_vector_type(2))) float v2f;
typedef __attribute__((ext_vector_type(8))) float v8f;
typedef __attribute__((vector_size(16)))    int   vi4;   // b128 transfer type

#if defined(__has_builtin)
#if __has_builtin(__builtin_amdgcn_global_load_async_to_lds_b128) && \
    __has_builtin(__builtin_amdgcn_s_wait_asynccnt)
#define USE_ASYNC_LDS 1
#endif
#endif

// ---------------------------------------------------------------------------
// Kernel 1: mid[M][16] = SCALING * (x @ A^T).  One wave per 16 rows.
// Per lane: A-frag element (M = lane&15, K = 2*half+j) and B-frag element
// (K = 2*half+j, N = lane&15) -> one global_load_b64 per fragment per step.
// ---------------------------------------------------------------------------
__global__ void __launch_bounds__(256)
lora_mid_kernel(const float* __restrict__ x, const float* __restrict__ Amat,
                float* __restrict__ mid, int D) {
    const int lane = threadIdx.x & 31;
    const int wave = threadIdx.x >> 5;
    const int half = lane >> 4;
    const int q    = lane & 15;

    const int row_base = blockIdx.x * 128 + wave * 16;   // 8 waves * 16 rows
    const int m_row    = row_base + q;

    v8f acc = {};

    // Lane-fixed base pointers: column offset 2*half selects this lane's pair.
    const float* xp = x    + (size_t)m_row * D + 2 * half;
    const float* ap = Amat + (size_t)q * D     + 2 * half;

    for (int k0 = 0; k0 < D; k0 += 16) {
        if (k0 + 16 < D) {
            __builtin_prefetch(xp + k0 + 16, 0, 3);      // global_prefetch_b8
            __builtin_prefetch(ap + k0 + 16, 0, 3);
        }
#pragma unroll
        for (int s = 0; s < 4; ++s) {
            const v2f av = *(const v2f*)(xp + k0 + 4 * s);
            const v2f bv = *(const v2f*)(ap + k0 + 4 * s);
            acc = __builtin_amdgcn_wmma_f32_16x16x4_f32(
                false, av, false, bv, (short)0, acc, false, false);
        }
    }

    // C/D layout: vgpr i -> M = i + 8*half, N = q.  Fold SCALING = 2.0 here.
#pragma unroll
    for (int i = 0; i < 8; ++i) {
        mid[(size_t)(row_base + i + 8 * half) * 16 + q] = 2.0f * acc[i];
    }
}

// ---------------------------------------------------------------------------
// Kernel 2: out = x @ W^T + mid @ B^T.
//   Block: 256 threads (8 waves), tile 128(M) x 64(N), K-step 16 via LDS.
//   Staging uses GLOBAL_LOAD_ASYNC_TO_LDS_B128 when available (ASYNCcnt),
//   else global->VGPR->ds_store fallback.
//   Wave grid 4x2; each wave computes a 32x32 tile = 2x2 WMMA accumulators.
//   LoRA rank-16 contribution fused into the epilogue (4 extra WMMAs / tile).
// ---------------------------------------------------------------------------
#define BM 128
#define BN 64
#define KT 16
#define LDP 20   // padded row stride in floats: 20*4 = 80 B, 16-B aligned

__global__ void __launch_bounds__(256)
lora_gemm_kernel(const float* __restrict__ x, const float* __restrict__ W,
                 const float* __restrict__ mid, const float* __restrict__ Bm,
                 float* __restrict__ out, int D, int O) {
    __shared__ float xs[BM][LDP];
    __shared__ float wsh[BN][LDP];

    const int tid  = threadIdx.x;
    const int lane = tid & 31;
    const int wave = tid >> 5;
    const int half = lane >> 4;
    const int q    = lane & 15;

    const int wm = wave >> 1;            // 0..3  -> M offset wm*32
    const int wn = wave & 1;             // 0..1  -> N offset wn*32

    const int m_block = blockIdx.x * BM;
    const int n_block = blockIdx.y * BN;

    // Tile-load assignments (one float4 per slot)
    const int xrow0 = (tid + 0)   >> 2, xkq0 = ((tid + 0)   & 3) * 4;
    const int xrow1 = (tid + 256) >> 2, xkq1 = ((tid + 256) & 3) * 4;
    const int wrow  = tid >> 2,         wkq  = (tid & 3) * 4;

    const float* xg0 = x + (size_t)(m_block + xrow0) * D + xkq0;
    const float* xg1 = x + (size_t)(m_block + xrow1) * D + xkq1;
    const float* wg  = W + (size_t)(n_block + wrow)  * D + wkq;

    v8f acc[2][2] = {};

    for (int k0 = 0; k0 < D; k0 += KT) {
#if defined(USE_ASYNC_LDS)
        // Async DMA global -> LDS (b128 per lane), tracked by ASYNCcnt.
        __builtin_amdgcn_global_load_async_to_lds_b128(
            (vi4*)(xg0 + k0), (vi4*)&xs[xrow0][xkq0], 0, 0);
        __builtin_amdgcn_global_load_async_to_lds_b128(
            (vi4*)(xg1 + k0), (vi4*)&xs[xrow1][xkq1], 0, 0);
        __builtin_amdgcn_global_load_async_to_lds_b128(
            (vi4*)(wg + k0),  (vi4*)&wsh[wrow][wkq],  0, 0);
#else
        *(float4*)&xs[xrow0][xkq0] = *(const float4*)(xg0 + k0);
        *(float4*)&xs[xrow1][xkq1] = *(const float4*)(xg1 + k0);
        *(float4*)&wsh[wrow][wkq]  = *(const float4*)(wg + k0);
#endif

        if (k0 + KT < D) {   // uniform branch: EXEC stays all-ones
            __builtin_prefetch(xg0 + k0 + KT, 0, 3);
            __builtin_prefetch(wg + k0 + KT, 0, 3);
        }

#if defined(USE_ASYNC_LDS)
        __builtin_amdgcn_s_wait_asynccnt(0);   // own async writes landed in LDS
#endif
        __syncthreads();

#pragma unroll
        for (int s = 0; s < 4; ++s) {
            const int koff = s * 4 + 2 * half;
            v2f av[2], bv[2];
#pragma unroll
            for (int i = 0; i < 2; ++i) {
                const int r = wm * 32 + i * 16 + q;
                av[i].x = xs[r][koff];
                av[i].y = xs[r][koff + 1];
            }
#pragma unroll
            for (int jn = 0; jn < 2; ++jn) {
                const int c = wn * 32 + jn * 16 + q;
                bv[jn].x = wsh[c][koff];
                bv[jn].y = wsh[c][koff + 1];
            }
#pragma unroll
            for (int i = 0; i < 2; ++i)
#pragma unroll
                for (int jn = 0; jn < 2; ++jn)
                    acc[i][jn] = __builtin_amdgcn_wmma_f32_16x16x4_f32(
                        false, av[i], false, bv[jn], (short)0, acc[i][jn],
                        false, false);
        }

        __syncthreads();
    }

    // LoRA epilogue: acc += mid(M x 16) @ B^T(16 x O) over rank K = 16.
    // mid already carries the 2.0 scaling.  Tiny operands -> L2 resident.
    const float* mp = mid + 2 * half;   // lane-fixed pair offset
    const float* bp = Bm  + 2 * half;
#pragma unroll
    for (int s = 0; s < 4; ++s) {
        v2f av[2], bv[2];
#pragma unroll
        for (int i = 0; i < 2; ++i) {
            const int m = m_block + wm * 32 + i * 16 + q;
            av[i] = *(const v2f*)(mp + (size_t)m * 16 + 4 * s);
        }
#pragma unroll
        for (int jn = 0; jn < 2; ++jn) {
            const int o = n_block + wn * 32 + jn * 16 + q;
            bv[jn] = *(const v2f*)(bp + (size_t)o * 16 + 4 * s);
        }
#pragma unroll
        for (int i = 0; i < 2; ++i)
#pragma unroll
            for (int jn = 0; jn < 2; ++jn)
                acc[i][jn] = __builtin_amdgcn_wmma_f32_16x16x4_f32(
                    false, av[i], false, bv[jn], (short)0, acc[i][jn],
                    false, false);
    }

    // Store: vgpr v -> (M = v + 8*half, N = q) within each 16x16 tile.
#pragma unroll
    for (int i = 0; i < 2; ++i)
#pragma unroll
        for (int jn = 0; jn < 2; ++jn) {
            const int col = n_block + wn * 32 + jn * 16 + q;
#pragma unroll
            for (int v = 0; v < 8; ++v) {
                const int row = m_block + wm * 32 + i * 16 + v + 8 * half;
                out[(size_t)row * O + col] = acc[i][jn][v];
            }
        }
}

extern "C" void kernel_launch(void* const* d_in, const int* in_sizes, int n_in,
                              void* d_out, int out_size, void* d_ws, size_t ws_size,
                              hipStream_t stream) {
    const float* x  = (const float*)d_in[0];   // [B,S,D] f32
    const float* W  = (const float*)d_in[1];   // [O,D]   f32
    const float* A  = (const float*)d_in[2];   // [16,D]  f32
    const float* Bm = (const float*)d_in[3];   // [O,16]  f32
    float* out = (float*)d_out;
    float* mid = (float*)d_ws;                 // [M,16] scratch, 512 KB

    const int D = in_sizes[2] / 16;            // 4096
    const int O = in_sizes[3] / 16;            // 4096
    const int M = (int)((size_t)in_sizes[0] / (size_t)D);  // 8192

    lora_mid_kernel<<<M / 128, 256, 0, stream>>>(x, A, mid, D);
    lora_gemm_kernel<<<dim3(M / BM, O / BN), 256, 0, stream>>>(x, W, mid, Bm, out, D, O);
}